// DP_88871463288893
// MI455X (gfx1250) — compile-verified
//
#include <hip/hip_runtime.h>
#include <hip/hip_bf16.h>

// ---------------------------------------------------------------------------
// DeepMD-style energy/force model on MI455X (gfx1250).
// All fp32 GEMMs are mapped to V_WMMA_F32_16X16X4_F32 (the only fp32 WMMA on
// CDNA5). Wave32: one 16-row tile per wave. LDS stages activations; weights
// stream from global (small, cache resident). Scatter uses f32 global atomics.
// ---------------------------------------------------------------------------

typedef float v2f __attribute__((ext_vector_type(2)));
typedef float v8f __attribute__((ext_vector_type(8)));

#define MAXN_ 128
#define NNEI 256          // NTYPES*MAXN
#define NATOM 4096
#define NAT_T 2048
#define E1 25
#define E2 50
#define E3 100
#define FIN 1600
#define FH 240
#define EMB_SCALE (1.0f/256.0f)   // 4/(MAXN*NTYPES*4)

struct EmbW  { const float *W1,*b1,*W2,*b2,*W3,*b3; };
struct EmbW4 { EmbW p[4]; };
struct FitW  { const float *W0,*b0,*W1,*b1,*W2,*b2,*W3,*b3; };
struct FitW2 { FitW p[2]; };

__device__ __forceinline__ v8f vzero8() { v8f z; for (int i = 0; i < 8; ++i) z[i] = 0.f; return z; }

__device__ __forceinline__ v8f wmma4(v2f a, v2f b, v8f c) {
  // D = A(16x4) x B(4x16) + C(16x16), fp32
  return __builtin_amdgcn_wmma_f32_16x16x4_f32(false, a, false, b, (short)0, c, false, false);
}

// A fragment (16x4) from row-major [rows x ld]; K columns [k0,k0+4), zero pad k>=kmax
__device__ __forceinline__ v2f ldA(const float* base, int ld, int r, int half, int k0, int kmax) {
  int k = k0 + 2*half;
  v2f a;
  a.x = (k     < kmax) ? base[r*ld + k    ] : 0.f;
  a.y = (k + 1 < kmax) ? base[r*ld + k + 1] : 0.f;
  return a;
}
// B fragment (4x16) from row-major weight W[din x dout]: B[kk][n] = W[k0+kk][n]
__device__ __forceinline__ v2f ldB(const float* W, int dout, int k0, int half, int n, int kmax, int nmax) {
  int k = k0 + 2*half;
  bool nv = n < nmax;
  v2f b;
  b.x = (nv && k     < kmax) ? W[(size_t)(k    )*dout + n] : 0.f;
  b.y = (nv && k + 1 < kmax) ? W[(size_t)(k + 1)*dout + n] : 0.f;
  return b;
}
// B = W^T: W row-major [nmax x ldW]: B[kk][n] = W[n][k0+kk]
__device__ __forceinline__ v2f ldBT(const float* W, int ldW, int k0, int half, int n, int kmax, int nmax) {
  int k = k0 + 2*half;
  bool nv = n < nmax;
  v2f b;
  b.x = (nv && k     < kmax) ? W[(size_t)n*ldW + k    ] : 0.f;
  b.y = (nv && k + 1 < kmax) ? W[(size_t)n*ldW + k + 1] : 0.f;
  return b;
}

// ---------------------------------------------------------------------------
// Shared embed forward over one chunk of 16 neighbors (recomputed in bwd too).
// h1: tanh(s*W1+b1)      (16x25, pad 28)
// h2: tanh(h1@W2+b2)+concat(h1,h1)   (16x50, pad 64)
// h3: tanh(h2@W3+b3)+concat(h2,h2)   (16x100, pad 112)
// ---------------------------------------------------------------------------
__device__ void embed_fwd_chunk(const float* RiA, const EmbW& w, int nb0,
                                int lane, int half, int r,
                                float (*h1s)[28], float (*h2s)[64], float (*h3s)[112]) {
  for (int idx = lane; idx < 16*28; idx += 32) {
    int k = idx / 28, j = idx - k*28;
    float v = 0.f;
    if (j < E1) {
      float s = RiA[(nb0 + k)*4 + 0];
      v = tanhf(s * w.W1[j] + w.b1[j]);
    }
    h1s[k][j] = v;
  }
  __syncthreads();
  for (int nt = 0; nt < 4; ++nt) {            // 25 -> 50
    int col = nt*16 + r;
    v8f c = vzero8();
    for (int kc = 0; kc < 7; ++kc) {
      v2f a = ldA(&h1s[0][0], 28, r, half, kc*4, E1);
      v2f b = ldB(w.W2, E2, kc*4, half, col, E1, E2);
      c = wmma4(a, b, c);
    }
    for (int j = 0; j < 8; ++j) {
      int k = half*8 + j;
      float v = 0.f;
      if (col < E2) v = tanhf(c[j] + w.b2[col]) + h1s[k][col % E1];
      h2s[k][col] = v;
    }
  }
  __syncthreads();
  for (int nt = 0; nt < 7; ++nt) {            // 50 -> 100
    int col = nt*16 + r;
    v8f c = vzero8();
    for (int kc = 0; kc < 13; ++kc) {
      v2f a = ldA(&h2s[0][0], 64, r, half, kc*4, E2);
      v2f b = ldB(w.W3, E3, kc*4, half, col, E2, E3);
      c = wmma4(a, b, c);
    }
    for (int j = 0; j < 8; ++j) {
      int k = half*8 + j;
      float v = 0.f;
      if (col < E3) v = tanhf(c[j] + w.b3[col]) + h2s[k][col % E2];
      h3s[k][col] = v;
    }
  }
  __syncthreads();
}

// ---------------------------------------------------------------------------
// K1: embed forward + A_{c,m} = scale * sum_{t1,k} blk[k][c]*G[k][m]  (4x100/atom)
// one wave per atom
// ---------------------------------------------------------------------------
__global__ __launch_bounds__(32) void k_embed_fwd(const float* __restrict__ Ri, EmbW4 ew,
                                                  float* __restrict__ Abuf) {
  __shared__ float h1s[16][28];
  __shared__ float h2s[16][64];
  __shared__ float h3s[16][112];
  int atom = blockIdx.x;
  int t = atom >= NAT_T;
  int lane = threadIdx.x & 31, half = lane >> 4, r = lane & 15;
  const float* RiA = Ri + (size_t)atom * NNEI * 4;
  v8f acc[7];
  for (int i = 0; i < 7; ++i) acc[i] = vzero8();
  for (int t1 = 0; t1 < 2; ++t1) {
    EmbW w = ew.p[t*2 + t1];
    for (int ch = 0; ch < 8; ++ch) {
      int nb0 = t1*MAXN_ + ch*16;
      embed_fwd_chunk(RiA, w, nb0, lane, half, r, h1s, h2s, h3s);
      // C_{c,m} += blk^T(4x16) @ G(16x100): M=c(4 of 16), K=neighbors, N=m
      for (int nt = 0; nt < 7; ++nt) {
        for (int kc = 0; kc < 4; ++kc) {
          int kk = kc*4 + 2*half;
          v2f a, b;
          a.x = (r < 4) ? RiA[(nb0 + kk    )*4 + r] : 0.f;
          a.y = (r < 4) ? RiA[(nb0 + kk + 1)*4 + r] : 0.f;
          b.x = h3s[kk    ][nt*16 + r];
          b.y = h3s[kk + 1][nt*16 + r];
          acc[nt] = wmma4(a, b, acc[nt]);
        }
      }
      __syncthreads();
    }
  }
  for (int nt = 0; nt < 7; ++nt) {
    int col = nt*16 + r;
    for (int j = 0; j < 8; ++j) {
      int row = half*8 + j;
      if (row < 4 && col < E3)
        Abuf[(size_t)atom*400 + row*100 + col] = acc[nt][j] * EMB_SCALE;
    }
  }
}

// K2: DR[m][k] = sum_c A[c][m]*A[c][k], k<16  (1600/atom)
__global__ __launch_bounds__(256) void k_dr(const float* __restrict__ Abuf, float* __restrict__ DR) {
  int atom = blockIdx.x;
  const float* A = Abuf + (size_t)atom*400;
  float* d = DR + (size_t)atom*FIN;
  for (int idx = threadIdx.x; idx < FIN; idx += blockDim.x) {
    int m = idx >> 4, k = idx & 15;
    float s = 0.f;
    for (int c = 0; c < 4; ++c) s += A[c*100 + m] * A[c*100 + k];
    d[idx] = s;
  }
}

// K3a: H1 = tanh(DR @ W0 + b0)   (4096x1600)@(1600x240)
__global__ __launch_bounds__(128) void k_fit_l1(const float* __restrict__ X, FitW2 fw,
                                                float* __restrict__ H1) {
  int wave = blockIdx.x * (blockDim.x >> 5) + (threadIdx.x >> 5);
  int row0 = wave * 16;
  if (row0 >= NATOM) return;
  FitW w = fw.p[row0 >= NAT_T];
  int lane = threadIdx.x & 31, half = lane >> 4, r = lane & 15;
  const float* Xr = X + (size_t)row0 * FIN;
  for (int nt = 0; nt < FH/16; ++nt) {
    int col = nt*16 + r;
    v8f c = vzero8();
    for (int kc = 0; kc < FIN/4; ++kc) {
      int k = kc*4 + 2*half;
      v2f a, b;
      a.x = Xr[(size_t)r*FIN + k];
      a.y = Xr[(size_t)r*FIN + k + 1];
      b.x = w.W0[(size_t)k*FH + col];
      b.y = w.W0[(size_t)(k + 1)*FH + col];
      c = wmma4(a, b, c);
    }
    for (int j = 0; j < 8; ++j) {
      int row = row0 + half*8 + j;
      H1[(size_t)row*FH + col] = tanhf(c[j] + w.b0[col]);
    }
  }
}

// K3b: Hout = tanh(Hin @ W + b) + Hin   (residual 240x240), layer = 1 or 2
__global__ __launch_bounds__(128) void k_fit_res(const float* __restrict__ Hin, FitW2 fw,
                                                 int layer, float* __restrict__ Hout) {
  int wave = blockIdx.x * (blockDim.x >> 5) + (threadIdx.x >> 5);
  int row0 = wave * 16;
  if (row0 >= NATOM) return;
  FitW w = fw.p[row0 >= NAT_T];
  const float* W = (layer == 1) ? w.W1 : w.W2;
  const float* bb = (layer == 1) ? w.b1 : w.b2;
  int lane = threadIdx.x & 31, half = lane >> 4, r = lane & 15;
  for (int nt = 0; nt < FH/16; ++nt) {
    int col = nt*16 + r;
    v8f c = vzero8();
    for (int kc = 0; kc < FH/4; ++kc) {
      int k = kc*4 + 2*half;
      v2f a, b;
      a.x = Hin[(size_t)(row0 + r)*FH + k];
      a.y = Hin[(size_t)(row0 + r)*FH + k + 1];
      b.x = W[(size_t)k*FH + col];
      b.y = W[(size_t)(k + 1)*FH + col];
      c = wmma4(a, b, c);
    }
    for (int j = 0; j < 8; ++j) {
      int row = row0 + half*8 + j;
      Hout[(size_t)row*FH + col] = tanhf(c[j] + bb[col]) + Hin[(size_t)row*FH + col];
    }
  }
}

// K3c: Ei = H3 . W3 + b3; Etot += Ei (atomic). One wave per atom.
__global__ __launch_bounds__(256) void k_ei(const float* __restrict__ H3, FitW2 fw,
                                            float* __restrict__ out) {
  int atom = blockIdx.x * (blockDim.x >> 5) + (threadIdx.x >> 5);
  if (atom >= NATOM) return;
  FitW w = fw.p[atom >= NAT_T];
  int lane = threadIdx.x & 31;
  float s = 0.f;
  for (int j = lane; j < FH; j += 32) s += H3[(size_t)atom*FH + j] * w.W3[j];
  for (int off = 16; off; off >>= 1) s += __shfl_down(s, off, 32);
  if (lane == 0) {
    float ei = s + w.b3[0];
    out[1 + atom] = ei;          // Ei
    atomicAdd(out, ei);          // Etot
  }
}

// K4a: gz3 = W3 * (1 - tanh(z3)^2), tanh(z3) = h3 - h2
__global__ __launch_bounds__(256) void k_gz3(const float* __restrict__ H2, const float* __restrict__ H3,
                                             FitW2 fw, float* __restrict__ GZ3) {
  size_t i = (size_t)blockIdx.x*blockDim.x + threadIdx.x;
  if (i >= (size_t)NATOM*FH) return;
  int atom = (int)(i / FH), j = (int)(i % FH);
  FitW w = fw.p[atom >= NAT_T];
  float tt = H3[i] - H2[i];
  GZ3[i] = w.W3[j] * (1.f - tt*tt);
}

// K4b: gh2 = gz3 @ W2^T + W3;  gz2 = gh2 * (1 - (h2-h1)^2)
__global__ __launch_bounds__(128) void k_bwd2(const float* __restrict__ GZ3,
                                              const float* __restrict__ H1, const float* __restrict__ H2,
                                              FitW2 fw, float* __restrict__ GH2, float* __restrict__ GZ2) {
  int wave = blockIdx.x * (blockDim.x >> 5) + (threadIdx.x >> 5);
  int row0 = wave * 16;
  if (row0 >= NATOM) return;
  FitW w = fw.p[row0 >= NAT_T];
  int lane = threadIdx.x & 31, half = lane >> 4, r = lane & 15;
  for (int nt = 0; nt < FH/16; ++nt) {
    int col = nt*16 + r;
    v8f c = vzero8();
    for (int kc = 0; kc < FH/4; ++kc) {
      int k = kc*4 + 2*half;
      v2f a;
      a.x = GZ3[(size_t)(row0 + r)*FH + k];
      a.y = GZ3[(size_t)(row0 + r)*FH + k + 1];
      v2f b = ldBT(w.W2, FH, kc*4, half, col, FH, FH);
      c = wmma4(a, b, c);
    }
    for (int j = 0; j < 8; ++j) {
      int row = row0 + half*8 + j;
      float gh2 = c[j] + w.W3[col];
      float tt = H2[(size_t)row*FH + col] - H1[(size_t)row*FH + col];
      GH2[(size_t)row*FH + col] = gh2;
      GZ2[(size_t)row*FH + col] = gh2 * (1.f - tt*tt);
    }
  }
}

// K4c: gh1 = gz2 @ W1^T + gh2;  gz1 = gh1 * (1 - h1^2)
__global__ __launch_bounds__(128) void k_bwd1(const float* __restrict__ GZ2,
                                              const float* __restrict__ GH2, const float* __restrict__ H1,
                                              FitW2 fw, float* __restrict__ GZ1) {
  int wave = blockIdx.x * (blockDim.x >> 5) + (threadIdx.x >> 5);
  int row0 = wave * 16;
  if (row0 >= NATOM) return;
  FitW w = fw.p[row0 >= NAT_T];
  int lane = threadIdx.x & 31, half = lane >> 4, r = lane & 15;
  for (int nt = 0; nt < FH/16; ++nt) {
    int col = nt*16 + r;
    v8f c = vzero8();
    for (int kc = 0; kc < FH/4; ++kc) {
      int k = kc*4 + 2*half;
      v2f a;
      a.x = GZ2[(size_t)(row0 + r)*FH + k];
      a.y = GZ2[(size_t)(row0 + r)*FH + k + 1];
      v2f b = ldBT(w.W1, FH, kc*4, half, col, FH, FH);
      c = wmma4(a, b, c);
    }
    for (int j = 0; j < 8; ++j) {
      int row = row0 + half*8 + j;
      float gh1 = c[j] + GH2[(size_t)row*FH + col];
      float tt = H1[(size_t)row*FH + col];
      GZ1[(size_t)row*FH + col] = gh1 * (1.f - tt*tt);
    }
  }
}

// K4d: gDR = gz1 @ W0^T   (4096x240)@(240x1600)
__global__ __launch_bounds__(128) void k_gdr(const float* __restrict__ GZ1, FitW2 fw,
                                             float* __restrict__ GDR) {
  int wave = blockIdx.x * (blockDim.x >> 5) + (threadIdx.x >> 5);
  int row0 = wave * 16;
  if (row0 >= NATOM) return;
  FitW w = fw.p[row0 >= NAT_T];
  int lane = threadIdx.x & 31, half = lane >> 4, r = lane & 15;
  for (int nt = 0; nt < FIN/16; ++nt) {
    int col = nt*16 + r;
    v8f c = vzero8();
    for (int kc = 0; kc < FH/4; ++kc) {
      int k = kc*4 + 2*half;
      v2f a;
      a.x = GZ1[(size_t)(row0 + r)*FH + k];
      a.y = GZ1[(size_t)(row0 + r)*FH + k + 1];
      v2f b = ldBT(w.W0, FH, kc*4, half, col, FH, FIN);
      c = wmma4(a, b, c);
    }
    for (int j = 0; j < 8; ++j) {
      int row = row0 + half*8 + j;
      GDR[(size_t)row*FIN + col] = c[j];
    }
  }
}

// K5: dT[c][m] = scale * ( sum_k gDR[m][k]*A[c][k]  +  (m<16) sum_mm gDR[mm][m]*A[c][mm] )
__global__ __launch_bounds__(128) void k_dA(const float* __restrict__ Abuf,
                                            const float* __restrict__ GDR, float* __restrict__ DT) {
  int atom = blockIdx.x;
  const float* A = Abuf + (size_t)atom*400;
  const float* g = GDR + (size_t)atom*FIN;
  float* dt = DT + (size_t)atom*400;
  for (int idx = threadIdx.x; idx < 400; idx += blockDim.x) {
    int c = idx / 100, m = idx - c*100;
    float s = 0.f;
    for (int k = 0; k < 16; ++k) s += g[m*16 + k] * A[c*100 + k];
    if (m < 16)
      for (int mm = 0; mm < 100; ++mm) s += g[mm*16 + m] * A[c*100 + mm];
    dt[idx] = s * EMB_SCALE;
  }
}

// ---------------------------------------------------------------------------
// K6: embed backward -> dE(atom, nb, c). Recomputes forward activations.
// one wave per atom.
// ---------------------------------------------------------------------------
__global__ __launch_bounds__(32) void k_embed_bwd(const float* __restrict__ Ri, EmbW4 ew,
                                                  const float* __restrict__ DT,
                                                  float* __restrict__ dE) {
  __shared__ float h1s[16][28];
  __shared__ float h2s[16][64];
  __shared__ float h3s[16][112];
  __shared__ float dGs[16][112];
  __shared__ float dz3s[16][112];
  __shared__ float dh2s[16][64];
  __shared__ float dz2s[16][64];
  __shared__ float dz1s[16][28];
  __shared__ float dts[4][112];
  __shared__ float dss[16];
  int atom = blockIdx.x;
  int t = atom >= NAT_T;
  int lane = threadIdx.x & 31, half = lane >> 4, r = lane & 15;
  const float* RiA = Ri + (size_t)atom * NNEI * 4;
  for (int idx = lane; idx < 4*112; idx += 32) {
    int c = idx / 112, m = idx - c*112;
    dts[c][m] = (m < E3) ? DT[(size_t)atom*400 + c*100 + m] : 0.f;
  }
  __syncthreads();
  for (int t1 = 0; t1 < 2; ++t1) {
    EmbW w = ew.p[t*2 + t1];
    for (int ch = 0; ch < 8; ++ch) {
      int nb0 = t1*MAXN_ + ch*16;
      embed_fwd_chunk(RiA, w, nb0, lane, half, r, h1s, h2s, h3s);
      // dG_{k,m} = sum_c blk[k][c]*dT[c][m]  (one K=4 chunk per N tile)
      for (int nt = 0; nt < 7; ++nt) {
        int col = nt*16 + r;
        v2f a, b;
        a.x = RiA[(nb0 + r)*4 + 2*half];
        a.y = RiA[(nb0 + r)*4 + 2*half + 1];
        b.x = dts[2*half    ][col];
        b.y = dts[2*half + 1][col];
        v8f c8 = wmma4(a, b, vzero8());
        for (int j = 0; j < 8; ++j) dGs[half*8 + j][col] = c8[j];
      }
      // dblk_{k,c} = sum_m G[k][m]*dT[c][m]
      v8f cb = vzero8();
      for (int kc = 0; kc < 25; ++kc) {
        v2f a = ldA(&h3s[0][0], 112, r, half, kc*4, E3);
        int k = kc*4 + 2*half;
        v2f b;
        b.x = (r < 4) ? dts[r][k    ] : 0.f;
        b.y = (r < 4) ? dts[r][k + 1] : 0.f;
        cb = wmma4(a, b, cb);
      }
      __syncthreads();
      // dz3 = dG * (1 - (h3 - concat(h2,h2))^2)
      for (int idx = lane; idx < 16*112; idx += 32) {
        int k = idx / 112, n = idx - k*112;
        float v = 0.f;
        if (n < E3) {
          float tt = h3s[k][n] - h2s[k][n % E2];
          v = dGs[k][n] * (1.f - tt*tt);
        }
        dz3s[k][n] = v;
      }
      __syncthreads();
      // dh2 = dz3 @ W3^T + dG[:,n] + dG[:,n+50]
      for (int nt = 0; nt < 4; ++nt) {
        int col = nt*16 + r;
        v8f c8 = vzero8();
        for (int kc = 0; kc < 25; ++kc) {
          v2f a = ldA(&dz3s[0][0], 112, r, half, kc*4, E3);
          v2f b = ldBT(w.W3, E3, kc*4, half, col, E3, E2);
          c8 = wmma4(a, b, c8);
        }
        for (int j = 0; j < 8; ++j) {
          int k = half*8 + j;
          float v = 0.f;
          if (col < E2) v = c8[j] + dGs[k][col] + dGs[k][col + E2];
          dh2s[k][col] = v;
        }
      }
      __syncthreads();
      // dz2 = dh2 * (1 - (h2 - concat(h1,h1))^2)
      for (int idx = lane; idx < 16*64; idx += 32) {
        int k = idx >> 6, n = idx & 63;
        float v = 0.f;
        if (n < E2) {
          float tt = h2s[k][n] - h1s[k][n % E1];
          v = dh2s[k][n] * (1.f - tt*tt);
        }
        dz2s[k][n] = v;
      }
      __syncthreads();
      // dh1 = dz2 @ W2^T + dh2[:,n] + dh2[:,n+25]; dz1 = dh1*(1-h1^2)
      for (int nt = 0; nt < 2; ++nt) {
        int col = nt*16 + r;
        v8f c8 = vzero8();
        for (int kc = 0; kc < 13; ++kc) {
          v2f a = ldA(&dz2s[0][0], 64, r, half, kc*4, E2);
          v2f b = ldBT(w.W2, E2, kc*4, half, col, E2, E1);
          c8 = wmma4(a, b, c8);
        }
        for (int j = 0; j < 8; ++j) {
          int k = half*8 + j;
          float v = 0.f;
          if (col < E1) {
            float gh1 = c8[j] + dh2s[k][col] + dh2s[k][col + E1];
            float tt = h1s[k][col];
            v = gh1 * (1.f - tt*tt);
          }
          if (col < 28) dz1s[k][col] = v;
        }
      }
      __syncthreads();
      // ds_k = sum_j dz1[k][j]*W1[j]  (layer-1 backprop, adds into coord 0)
      if (lane < 16) {
        float s = 0.f;
        for (int j = 0; j < E1; ++j) s += dz1s[lane][j] * w.W1[j];
        dss[lane] = s;
      }
      __syncthreads();
      // write dE = dblk (+ ds into c==0)
      {
        int col = r;
        if (col < 4) {
          for (int j = 0; j < 8; ++j) {
            int k = half*8 + j;
            float v = cb[j];
            if (col == 0) v += dss[k];
            dE[((size_t)atom*NNEI + nb0 + k)*4 + col] = v;
          }
        }
      }
      __syncthreads();
    }
  }
}

// K7a: F_n = -sum_p dE[n][p]*dfeat[n][p][:]. One wave per atom.
__global__ __launch_bounds__(256) void k_force_self(const float* __restrict__ dE,
                                                    const float* __restrict__ dfeat,
                                                    float* __restrict__ out) {
  int atom = blockIdx.x * (blockDim.x >> 5) + (threadIdx.x >> 5);
  if (atom >= NATOM) return;
  int lane = threadIdx.x & 31;
  const float* e = dE + (size_t)atom*1024;
  const float* df = dfeat + (size_t)atom*1024*3;
  float fx = 0.f, fy = 0.f, fz = 0.f;
  for (int p = lane; p < 1024; p += 32) {
    float g = e[p];
    fx += g*df[p*3 + 0]; fy += g*df[p*3 + 1]; fz += g*df[p*3 + 2];
  }
  for (int off = 16; off; off >>= 1) {
    fx += __shfl_down(fx, off, 32);
    fy += __shfl_down(fy, off, 32);
    fz += __shfl_down(fz, off, 32);
  }
  if (lane == 0) {
    float* F = out + 1 + NATOM;
    F[(size_t)atom*3 + 0] = -fx;
    F[(size_t)atom*3 + 1] = -fy;
    F[(size_t)atom*3 + 2] = -fz;
  }
}

// K7b: scatter contrib into F via f32 global atomics
__global__ __launch_bounds__(256) void k_scatter(const float* __restrict__ dE,
                                                 const float* __restrict__ dfeat,
                                                 const int* __restrict__ neigh,
                                                 float* __restrict__ out) {
  size_t i = (size_t)blockIdx.x*blockDim.x + threadIdx.x;  // (n,k) pair
  if (i >= (size_t)NATOM*NNEI) return;
  int idx = neigh[i] - 1;
  if (idx < 0 || idx >= NATOM) return;
  const float* e  = dE + i*4;
  const float* df = dfeat + i*12;
  float* F = out + 1 + NATOM;
  for (int d = 0; d < 3; ++d) {
    float c = e[0]*df[0*3 + d] + e[1]*df[1*3 + d] + e[2]*df[2*3 + d] + e[3]*df[3*3 + d];
    atomicAdd(&F[(size_t)idx*3 + d], c);
  }
}

__global__ void k_init(float* __restrict__ out) {
  if (threadIdx.x == 0 && blockIdx.x == 0) out[0] = 0.f;   // Etot accumulator
}

// ---------------------------------------------------------------------------
// Host launch. Input order (flattened leaves of setup_inputs()):
//  0 Ri, 1 dfeat, 2..25 emb_params (4 pairs x 3 layers x {W,b}),
//  26..41 fit_params (2 types x 4 layers x {W,b}), 42 list_neigh,
//  43 natoms_img, 44 Egroup_weight, 45 divider.
// Workspace: ~110 MB fp32.
// ---------------------------------------------------------------------------
extern "C" void kernel_launch(void* const* d_in, const int* in_sizes, int n_in,
                              void* d_out, int out_size, void* d_ws, size_t ws_size,
                              hipStream_t stream) {
  (void)in_sizes; (void)n_in; (void)out_size; (void)ws_size;
  const float* Ri    = (const float*)d_in[0];
  const float* dfeat = (const float*)d_in[1];
  EmbW4 ew;
  for (int p = 0; p < 4; ++p) {
    int b = 2 + p*6;
    ew.p[p].W1 = (const float*)d_in[b + 0];
    ew.p[p].b1 = (const float*)d_in[b + 1];
    ew.p[p].W2 = (const float*)d_in[b + 2];
    ew.p[p].b2 = (const float*)d_in[b + 3];
    ew.p[p].W3 = (const float*)d_in[b + 4];
    ew.p[p].b3 = (const float*)d_in[b + 5];
  }
  FitW2 fw;
  for (int t = 0; t < 2; ++t) {
    int b = 26 + t*8;
    fw.p[t].W0 = (const float*)d_in[b + 0];
    fw.p[t].b0 = (const float*)d_in[b + 1];
    fw.p[t].W1 = (const float*)d_in[b + 2];
    fw.p[t].b1 = (const float*)d_in[b + 3];
    fw.p[t].W2 = (const float*)d_in[b + 4];
    fw.p[t].b2 = (const float*)d_in[b + 5];
    fw.p[t].W3 = (const float*)d_in[b + 6];
    fw.p[t].b3 = (const float*)d_in[b + 7];
  }
  const int* neigh = (const int*)d_in[42];
  float* out = (float*)d_out;
  float* ws  = (float*)d_ws;

  size_t off = 0;
  float* Abuf = ws + off; off += (size_t)NATOM*400;
  float* DR   = ws + off; off += (size_t)NATOM*FIN;
  float* H1   = ws + off; off += (size_t)NATOM*FH;
  float* H2   = ws + off; off += (size_t)NATOM*FH;
  float* H3   = ws + off; off += (size_t)NATOM*FH;
  float* GZ3  = ws + off; off += (size_t)NATOM*FH;
  float* GH2  = ws + off; off += (size_t)NATOM*FH;
  float* GZ2  = ws + off; off += (size_t)NATOM*FH;
  float* GZ1  = ws + off; off += (size_t)NATOM*FH;
  float* GDR  = ws + off; off += (size_t)NATOM*FIN;
  float* DT   = ws + off; off += (size_t)NATOM*400;
  float* dEb  = ws + off; off += (size_t)NATOM*NNEI*4;

  hipLaunchKernelGGL(k_init,       dim3(1),          dim3(32),  0, stream, out);
  hipLaunchKernelGGL(k_embed_fwd,  dim3(NATOM),      dim3(32),  0, stream, Ri, ew, Abuf);
  hipLaunchKernelGGL(k_dr,         dim3(NATOM),      dim3(256), 0, stream, Abuf, DR);
  hipLaunchKernelGGL(k_fit_l1,     dim3(64),         dim3(128), 0, stream, DR, fw, H1);
  hipLaunchKernelGGL(k_fit_res,    dim3(64),         dim3(128), 0, stream, H1, fw, 1, H2);
  hipLaunchKernelGGL(k_fit_res,    dim3(64),         dim3(128), 0, stream, H2, fw, 2, H3);
  hipLaunchKernelGGL(k_ei,         dim3(512),        dim3(256), 0, stream, H3, fw, out);
  hipLaunchKernelGGL(k_gz3,        dim3((NATOM*FH + 255)/256), dim3(256), 0, stream, H2, H3, fw, GZ3);
  hipLaunchKernelGGL(k_bwd2,       dim3(64),         dim3(128), 0, stream, GZ3, H1, H2, fw, GH2, GZ2);
  hipLaunchKernelGGL(k_bwd1,       dim3(64),         dim3(128), 0, stream, GZ2, GH2, H1, fw, GZ1);
  hipLaunchKernelGGL(k_gdr,        dim3(64),         dim3(128), 0, stream, GZ1, fw, GDR);
  hipLaunchKernelGGL(k_dA,         dim3(NATOM),      dim3(128), 0, stream, Abuf, GDR, DT);
  hipLaunchKernelGGL(k_embed_bwd,  dim3(NATOM),      dim3(32),  0, stream, Ri, ew, DT, dEb);
  hipLaunchKernelGGL(k_force_self, dim3(NATOM/8),    dim3(256), 0, stream, dEb, dfeat, out);
  hipLaunchKernelGGL(k_scatter,    dim3((NATOM*NNEI)/256), dim3(256), 0, stream, dEb, dfeat, neigh, out);
}